// GCMCRating1_68049461838612
// MI455X (gfx1250) — compile-verified
//
#include <hip/hip_runtime.h>

typedef __attribute__((ext_vector_type(2))) float v2f;
typedef __attribute__((ext_vector_type(8))) float v8f;

#define D 64

__device__ __forceinline__ v8f wmma_f32_4(v2f a, v2f b, v8f c) {
  // V_WMMA_F32_16X16X4_F32: D(16x16 f32) = A(16x4 f32) * B(4x16 f32) + C
  return __builtin_amdgcn_wmma_f32_16x16x4_f32(false, a, false, b, (short)0, c,
                                               false, false);
}

// ---------------- embedding gather: out[n,:] = emb_a[ids_a[n],:] + emb_b[ids_b[n],:] ----
__global__ void gather_add_kernel(const int* __restrict__ ids_a,
                                  const int* __restrict__ ids_b,
                                  const float* __restrict__ emb_a,
                                  const float* __restrict__ emb_b,
                                  float* __restrict__ out, int n) {
  int t = blockIdx.x * blockDim.x + threadIdx.x;
  int node = t >> 5;
  int lane = t & 31;
  if (node >= n) return;
  size_t ra = (size_t)ids_a[node] * D;
  size_t rb = (size_t)ids_b[node] * D;
  float2 x = *(const float2*)(emb_a + ra + lane * 2);
  float2 y = *(const float2*)(emb_b + rb + lane * 2);
  float2 o = make_float2(x.x + y.x, x.y + y.y);
  *(float2*)(out + (size_t)node * D + lane * 2) = o;
}

// -------- edge scatter: A6[rating, dst, :] += h_src[src, :]; cnt[dst] += 1 -----------
// One wave per edge; L2-resident atomics (A6 + h_src fit in the 192MB L2).
__global__ void edge_scatter_kernel(const int* __restrict__ src_idx,
                                    const int* __restrict__ dst_idx,
                                    const int* __restrict__ rating,
                                    const float* __restrict__ h_src,
                                    float* __restrict__ A6,
                                    float* __restrict__ cnt,
                                    int num_edges, int n_dst) {
  int w = (blockIdx.x * blockDim.x + threadIdx.x) >> 5;
  int lane = threadIdx.x & 31;
  if (w >= num_edges) return;
  int s = src_idx[w];
  int d = dst_idx[w];
  int r = rating[w];
  float2 v = *(const float2*)(h_src + (size_t)s * D + lane * 2);
  float* p = A6 + ((size_t)r * n_dst + (size_t)d) * D + lane * 2;
  atomicAdd(p, v.x);
  atomicAdd(p + 1, v.y);
  if (lane == 0) atomicAdd(cnt + d, 1.0f);
}

// ---- fused: neigh = (sum_r A6[r] @ Wr[r]^T)/max(cnt,1)
//      h_new = relu([h_dst | neigh] @ Wlin^T + blin)
//      out   = h_new @ Wfin^T + bfin
// One wave computes a 16-row x 64-col output tile via v_wmma_f32_16x16x4_f32 chains.
__global__ void conv_fused_kernel(const float* __restrict__ A6,   // [6, n, 64]
                                  const float* __restrict__ cnt,  // [n]
                                  const float* __restrict__ hdst, // [n, 64]
                                  const float* __restrict__ Wr,   // [6, 64, 64]
                                  const float* __restrict__ Wlin, // [64, 128]
                                  const float* __restrict__ blin, // [64]
                                  const float* __restrict__ Wfin, // [64, 64]
                                  const float* __restrict__ bfin, // [64]
                                  float* __restrict__ out,        // [n, 64]
                                  int n) {
  __shared__ float lds[8][16 * 66];   // per-wave 16x64 staging tile, padded stride
  const int wave = threadIdx.x >> 5;
  const int lane = threadIdx.x & 31;
  const int hlf  = lane >> 4;         // K-half select for A/B fragments
  const int ln   = lane & 15;         // row (A) / col (B,C) within tile
  const int tile = blockIdx.x * 8 + wave;
  const int row0 = tile * 16;
  if (row0 >= n) return;              // uniform per wave: EXEC stays all-1 for WMMA

  float* L = lds[wave];
  const int am = min(row0 + ln, n - 1);  // clamped source row for A fragments
  const int kh = 2 * hlf;

  // ---------------- stage 1: bucket GEMMs ----------------
  v8f acc[4] = {};
  #pragma unroll 1
  for (int r = 0; r < 6; ++r) {
    const float* Ar = A6 + ((size_t)r * n + (size_t)am) * D;
    const float* W  = Wr + r * (D * D);
    #pragma unroll
    for (int k0 = 0; k0 < D; k0 += 4) {
      v2f a;
      a.x = Ar[k0 + kh];
      a.y = Ar[k0 + kh + 1];
      #pragma unroll
      for (int c = 0; c < 4; ++c) {
        const float* Wn = W + (c * 16 + ln) * D + k0 + kh;  // B[k][n] = Wr[n][k]
        v2f b; b.x = Wn[0]; b.y = Wn[1];
        acc[c] = wmma_f32_4(a, b, acc[c]);
      }
    }
  }

  // mean: divide rows by max(cnt,1)
  float inv[8];
  #pragma unroll
  for (int v = 0; v < 8; ++v) {
    int m = row0 + v + 8 * hlf;
    float cv = (m < n) ? cnt[m] : 1.0f;
    inv[v] = 1.0f / fmaxf(cv, 1.0f);
  }
  #pragma unroll
  for (int c = 0; c < 4; ++c)
    #pragma unroll
    for (int v = 0; v < 8; ++v)
      acc[c][v] *= inv[v];

  // stash neigh tile (C-layout -> row-major LDS) for re-read in A-layout
  #pragma unroll
  for (int c = 0; c < 4; ++c)
    #pragma unroll
    for (int v = 0; v < 8; ++v)
      L[(v + 8 * hlf) * 66 + c * 16 + ln] = acc[c][v];

  // ---------------- stage 2: [h_dst | neigh] @ Wlin^T + blin, relu ----------------
  v8f acc2[4] = {};
  const float* Hm = hdst + (size_t)am * D;
  #pragma unroll
  for (int k0 = 0; k0 < D; k0 += 4) {          // K = 0..63 from h_dst (global/L2)
    v2f a; a.x = Hm[k0 + kh]; a.y = Hm[k0 + kh + 1];
    #pragma unroll
    for (int c = 0; c < 4; ++c) {
      const float* Wn = Wlin + (c * 16 + ln) * (2 * D) + k0 + kh;
      v2f b; b.x = Wn[0]; b.y = Wn[1];
      acc2[c] = wmma_f32_4(a, b, acc2[c]);
    }
  }
  #pragma unroll
  for (int k0 = 0; k0 < D; k0 += 4) {          // K = 64..127 from neigh (LDS)
    v2f a; a.x = L[ln * 66 + k0 + kh]; a.y = L[ln * 66 + k0 + kh + 1];
    #pragma unroll
    for (int c = 0; c < 4; ++c) {
      const float* Wn = Wlin + (c * 16 + ln) * (2 * D) + D + k0 + kh;
      v2f b; b.x = Wn[0]; b.y = Wn[1];
      acc2[c] = wmma_f32_4(a, b, acc2[c]);
    }
  }
  #pragma unroll
  for (int c = 0; c < 4; ++c) {
    float bb = blin[c * 16 + ln];
    #pragma unroll
    for (int v = 0; v < 8; ++v)
      acc2[c][v] = fmaxf(acc2[c][v] + bb, 0.0f);
  }
  // overwrite LDS with h_new (in-order per-wave DS ops: prior reads already done)
  #pragma unroll
  for (int c = 0; c < 4; ++c)
    #pragma unroll
    for (int v = 0; v < 8; ++v)
      L[(v + 8 * hlf) * 66 + c * 16 + ln] = acc2[c][v];

  // ---------------- stage 3: h_new @ Wfin^T + bfin ----------------
  v8f acc3[4] = {};
  #pragma unroll
  for (int k0 = 0; k0 < D; k0 += 4) {
    v2f a; a.x = L[ln * 66 + k0 + kh]; a.y = L[ln * 66 + k0 + kh + 1];
    #pragma unroll
    for (int c = 0; c < 4; ++c) {
      const float* Wn = Wfin + (c * 16 + ln) * D + k0 + kh;
      v2f b; b.x = Wn[0]; b.y = Wn[1];
      acc3[c] = wmma_f32_4(a, b, acc3[c]);
    }
  }
  #pragma unroll
  for (int c = 0; c < 4; ++c) {
    float bb = bfin[c * 16 + ln];
    #pragma unroll
    for (int v = 0; v < 8; ++v) {
      int m = row0 + v + 8 * hlf;
      if (m < n) out[(size_t)m * D + c * 16 + ln] = acc3[c][v] + bb;
    }
  }
}

extern "C" void kernel_launch(void* const* d_in, const int* in_sizes, int n_in,
                              void* d_out, int out_size, void* d_ws, size_t ws_size,
                              hipStream_t stream) {
  const int NU = in_sizes[0];
  const int NI = in_sizes[1];
  const int E  = in_sizes[4];

  const int*   user_ids     = (const int*)d_in[0];
  const int*   item_ids     = (const int*)d_in[1];
  const int*   gender       = (const int*)d_in[2];
  const int*   genres       = (const int*)d_in[3];
  const int*   edge_user    = (const int*)d_in[4];
  const int*   edge_item    = (const int*)d_in[5];
  const int*   rating       = (const int*)d_in[6];
  const float* user_emb     = (const float*)d_in[7];
  const float* item_emb     = (const float*)d_in[8];
  const float* gender_emb   = (const float*)d_in[9];
  const float* genre_emb    = (const float*)d_in[10];
  const float* Wr_watched   = (const float*)d_in[11];
  const float* Wr_watchedby = (const float*)d_in[12];
  const float* Ww  = (const float*)d_in[13];
  const float* bw  = (const float*)d_in[14];
  const float* Wwb = (const float*)d_in[15];
  const float* bwb = (const float*)d_in[16];
  const float* Wf  = (const float*)d_in[17];
  const float* bf  = (const float*)d_in[18];
  const float* Vf  = (const float*)d_in[19];
  const float* bv  = (const float*)d_in[20];

  // workspace: hu | hi | A_watched[6,NI,64] | cnt_i | A_watchedby[6,NU,64] | cnt_u
  float* ws    = (float*)d_ws;
  float* hu    = ws;
  float* hi    = hu + (size_t)NU * D;
  float* Aw    = hi + (size_t)NI * D;
  float* cnt_i = Aw + (size_t)6 * NI * D;
  float* Awb   = cnt_i + (size_t)NI;
  float* cnt_u = Awb + (size_t)6 * NU * D;

  size_t zero_bytes = ((size_t)6 * NI * D + NI + (size_t)6 * NU * D + NU) * sizeof(float);
  hipMemsetAsync(Aw, 0, zero_bytes, stream);

  // embeddings
  {
    long tu = (long)NU * 32;
    gather_add_kernel<<<(unsigned)((tu + 255) / 256), 256, 0, stream>>>(
        user_ids, gender, user_emb, gender_emb, hu, NU);
    long ti = (long)NI * 32;
    gather_add_kernel<<<(unsigned)((ti + 255) / 256), 256, 0, stream>>>(
        item_ids, genres, item_emb, genre_emb, hi, NI);
  }

  // edge accumulation (both directions)
  {
    unsigned blocks = (unsigned)((E + 7) / 8);   // 8 waves/block, 1 edge/wave
    edge_scatter_kernel<<<blocks, 256, 0, stream>>>(edge_user, edge_item, rating,
                                                    hu, Aw, cnt_i, E, NI);
    edge_scatter_kernel<<<blocks, 256, 0, stream>>>(edge_item, edge_user, rating,
                                                    hi, Awb, cnt_u, E, NU);
  }

  // fused conv + output projection; d_out = [user_out (NU*64) | item_out (NI*64)]
  {
    int tiles_i = (NI + 15) / 16;
    conv_fused_kernel<<<(unsigned)((tiles_i + 7) / 8), 256, 0, stream>>>(
        Aw, cnt_i, hi, Wr_watched, Ww, bw, Vf, bv,
        (float*)d_out + (size_t)NU * D, NI);
    int tiles_u = (NU + 15) / 16;
    conv_fused_kernel<<<(unsigned)((tiles_u + 7) / 8), 256, 0, stream>>>(
        Awb, cnt_u, hu, Wr_watchedby, Wwb, bwb, Wf, bf,
        (float*)d_out, NU);
  }
}